// ToBEVReduction_42279658062067
// MI455X (gfx1250) — compile-verified
//
#include <hip/hip_runtime.h>

// Problem constants (from the reference)
#define BDIM 8
#define YDIM 512
#define ZDIM 64
#define NUM_CELLS (BDIM * YDIM * ZDIM)   // 262144
#define CCH 128                          // channels
#define LANES 32                         // wave32: one wave handles one cell row
#define WAVES_PER_BLOCK 8                // 256 threads

typedef float  v4f __attribute__((ext_vector_type(4)));
typedef int    v4i __attribute__((ext_vector_type(4)));

// ---------------------------------------------------------------------------
// Kernel 1: head[cell] = -1 for all cells (vectorized 16B stores)
// ---------------------------------------------------------------------------
__global__ void __launch_bounds__(256)
tobev_init_heads(int* __restrict__ head) {
    int i = (blockIdx.x * blockDim.x + threadIdx.x) * 4;
    if (i < NUM_CELLS) {
        v4i m1 = {-1, -1, -1, -1};
        *(v4i*)(head + i) = m1;
    }
}

// ---------------------------------------------------------------------------
// Kernel 2: build per-cell linked lists with a single 32-bit swap atomic
// per point (600K atomics total instead of 76.8M float-add atomics).
// coords is (N,4) int32 row-major -> one coalesced b128 load per point.
// ---------------------------------------------------------------------------
__global__ void __launch_bounds__(256)
tobev_build_lists(const v4i* __restrict__ coords,
                  int* __restrict__ head,
                  int* __restrict__ next,
                  int n) {
    int i = blockIdx.x * blockDim.x + threadIdx.x;
    if (i >= n) return;
    v4i c = coords[i];                       // {b, x, y, z}; x (DIM=1) is dropped
    int cell = (c.x * YDIM + c.z) * ZDIM + c.w;
    int prev = atomicExch(&head[cell], i);   // global_atomic_swap_b32
    next[i] = prev;
}

// ---------------------------------------------------------------------------
// Kernel 3: one wave32 per cell; each lane owns 4 channels (float4), so the
// 32 lanes cover the full 512B feature row with one coalesced b128 load per
// point and one coalesced b128 store per cell.
//
// CDNA5 specifics:
//  * the pointer chase (head/next) is forced uniform via readfirstlane so it
//    lowers to s_load_b32 on the scalar pipe (KMcnt) and a pure-SALU loop —
//    the VMEM pipe only ever sees the big row transfers;
//  * feature rows are loaded GVS-style (SGPR base + per-lane VGPR offset)
//    with TH=NT (read-once stream must not evict head/next from the 192MB L2);
//  * global_prefetch_b8 pulls the next chain element's row while the VALU
//    accumulates the current one (f64 accumulation: order-robust, free at
//    this arithmetic intensity).
// ---------------------------------------------------------------------------
__global__ void __launch_bounds__(256)
tobev_gather_mean(const float* __restrict__ feats,
                  const int* __restrict__ head,
                  const int* __restrict__ next,
                  float* __restrict__ out) {
    const int lane = threadIdx.x & (LANES - 1);
    // Force wave-uniformity so the chain walk runs on the scalar pipe.
    const int wv   = __builtin_amdgcn_readfirstlane((int)(threadIdx.x >> 5));
    const int cell = (int)blockIdx.x * WAVES_PER_BLOCK + wv;   // uniform (SGPR)
    if (cell >= NUM_CELLS) return;

    int p = __builtin_amdgcn_readfirstlane(head[cell]);        // s_load path
    double a0 = 0.0, a1 = 0.0, a2 = 0.0, a3 = 0.0;
    int cnt = 0;

    while (p >= 0) {                                           // scalar branch
        int pn = __builtin_amdgcn_readfirstlane(next[p]);      // s_load path
        const v4f* row = (const v4f*)(feats + (size_t)p * CCH) + lane;
        if (pn >= 0) {
            const v4f* nrow = (const v4f*)(feats + (size_t)pn * CCH) + lane;
            __builtin_prefetch((const void*)nrow, 0, 1);       // global_prefetch_b8
        }
        v4f f = __builtin_nontemporal_load(row);               // b128, TH=NT
        a0 += (double)f.x;
        a1 += (double)f.y;
        a2 += (double)f.z;
        a3 += (double)f.w;
        cnt++;
        p = pn;
    }

    double d = (cnt > 0) ? (double)cnt : 1.0;
    v4f r;
    r.x = (float)(a0 / d);
    r.y = (float)(a1 / d);
    r.z = (float)(a2 / d);
    r.w = (float)(a3 / d);

    v4f* dst = (v4f*)(out + (size_t)cell * CCH) + lane;
    __builtin_nontemporal_store(r, dst);                       // b128, TH=NT
}

// ---------------------------------------------------------------------------
// Launch: init heads -> build lists -> gather means. All on `stream`,
// no allocation, no sync: graph-capture safe and deterministic in work.
// Workspace layout: [ head : NUM_CELLS ints ][ next : N ints ]  (~3.3 MB)
// ---------------------------------------------------------------------------
extern "C" void kernel_launch(void* const* d_in, const int* in_sizes, int n_in,
                              void* d_out, int out_size, void* d_ws, size_t ws_size,
                              hipStream_t stream) {
    const v4i*  coords = (const v4i*)d_in[0];    // (N,4) int32
    const float* feats = (const float*)d_in[1];  // (N,128) float32
    float* out = (float*)d_out;                  // (NUM_CELLS,128) float32

    const int n = in_sizes[0] / 4;               // N points

    int* head = (int*)d_ws;
    int* next = head + NUM_CELLS;

    // 1) reset heads (NUM_CELLS/4 threads, 16B stores)
    {
        int threads = NUM_CELLS / 4;
        tobev_init_heads<<<threads / 256, 256, 0, stream>>>(head);
    }
    // 2) build linked lists (one swap atomic per point)
    {
        int blocks = (n + 255) / 256;
        tobev_build_lists<<<blocks, 256, 0, stream>>>(coords, head, next, n);
    }
    // 3) wave-per-cell gather + mean (collision-free, bandwidth-bound)
    {
        int blocks = NUM_CELLS / WAVES_PER_BLOCK;
        tobev_gather_mean<<<blocks, 256, 0, stream>>>(feats, head, next, out);
    }
}